// LocallyConnected2d_55645596287382
// MI455X (gfx1250) — compile-verified
//
#include <hip/hip_runtime.h>

// LocallyConnected2d on MI455X (gfx1250):
//   per-location GEMM  M=32(batch) x N=32(outch) x K=144(cin*9)
//   fp32 throughout -> V_WMMA_F32_16X16X4_F32, 2x2 tiles of 16x16, K chunked by 4.
// Memory-bound (~102 MB traffic, ~1.2 GFLOP). LDS staging fixes the scattered
// weight layout [o][c][hw][9] into a [K][o] image readable in WMMA B striping.

typedef __attribute__((ext_vector_type(2))) float v2f;
typedef __attribute__((ext_vector_type(8))) float v8f;

#define KTOT 144  // 16 input channels * 9 kernel taps

__global__ __launch_bounds__(32) void lc2d_wmma_f32_kernel(
    const float* __restrict__ x,     // [32][16][66][66]
    const float* __restrict__ wgt,   // [32][16][64*64][9]
    const float* __restrict__ bias,  // [32][64*64]
    float* __restrict__ out)         // [32][32][64*64]
{
  __shared__ float lds_x[KTOT * 32];  // [K][b]  (A image)
  __shared__ float lds_w[KTOT * 32];  // [K][o]  (B image)

  const int lane = threadIdx.x;   // 0..31 (wave32)
  const int loc  = blockIdx.x;    // h*64 + w
  const int h = loc >> 6;
  const int w = loc & 63;

  // ---- stage weights: lane <-> output channel o; 16 runs of 9 contiguous floats
  {
    const int o = lane;
    #pragma unroll
    for (int c = 0; c < 16; ++c) {
      const float* src = wgt + (size_t)((o * 16 + c) * 4096 + loc) * 9;
      #pragma unroll
      for (int kk = 0; kk < 9; ++kk)
        lds_w[(c * 9 + kk) * 32 + o] = src[kk];
    }
  }

  // ---- stage x patches: lane <-> batch b; 48 runs of 3 contiguous floats
  {
    const int b = lane;
    #pragma unroll
    for (int c = 0; c < 16; ++c) {
      const float* src = x + ((size_t)(b * 16 + c) * 66 + h) * 66 + w;
      #pragma unroll
      for (int i = 0; i < 3; ++i)
        #pragma unroll
        for (int j = 0; j < 3; ++j)
          lds_x[(c * 9 + i * 3 + j) * 32 + b] = src[i * 66 + j];
    }
  }
  __syncthreads();

  const int t = lane >> 4;   // half-wave selector (K+0/K+1 vs K+2/K+3 striping)
  const int r = lane & 15;   // row/col within 16-wide tile

  v8f acc00 = {}, acc01 = {}, acc10 = {}, acc11 = {};

  #pragma unroll
  for (int kc = 0; kc < KTOT; kc += 4) {
    const int kA = kc + 2 * t;       // this half-wave's first K
    const int base = kA * 32;
    v2f a0, a1, b0, b1;
    // A fragments: VGPR0 = K(kA), VGPR1 = K(kA+1); M = r (+16 for tile m=1)
    a0.x = lds_x[base + r];           a0.y = lds_x[base + 32 + r];
    a1.x = lds_x[base + 16 + r];      a1.y = lds_x[base + 48 + r];
    // B fragments: same striping; N = r (+16 for tile n=1)
    b0.x = lds_w[base + r];           b0.y = lds_w[base + 32 + r];
    b1.x = lds_w[base + 16 + r];      b1.y = lds_w[base + 48 + r];

    acc00 = __builtin_amdgcn_wmma_f32_16x16x4_f32(false, a0, false, b0,
                                                  (short)0, acc00, false, false);
    acc01 = __builtin_amdgcn_wmma_f32_16x16x4_f32(false, a0, false, b1,
                                                  (short)0, acc01, false, false);
    acc10 = __builtin_amdgcn_wmma_f32_16x16x4_f32(false, a1, false, b0,
                                                  (short)0, acc10, false, false);
    acc11 = __builtin_amdgcn_wmma_f32_16x16x4_f32(false, a1, false, b1,
                                                  (short)0, acc11, false, false);
  }

  // ---- epilogue: bias add + store. C/D layout: VGPR row v holds M = v (lanes
  // 0-15) and M = v+8 (lanes 16-31); N = lane&15.
  const float bv0 = bias[(size_t)r * 4096 + loc];         // o = r       (n=0)
  const float bv1 = bias[(size_t)(r + 16) * 4096 + loc];  // o = r + 16  (n=1)

  #pragma unroll
  for (int row = 0; row < 8; ++row) {
    const int bm0 = row + 8 * t;        // batch for m=0 tile
    const int bm1 = 16 + row + 8 * t;   // batch for m=1 tile
    out[(size_t)(bm0 * 32 + r)        * 4096 + loc] = acc00[row] + bv0;
    out[(size_t)(bm0 * 32 + r + 16)   * 4096 + loc] = acc01[row] + bv1;
    out[(size_t)(bm1 * 32 + r)        * 4096 + loc] = acc10[row] + bv0;
    out[(size_t)(bm1 * 32 + r + 16)   * 4096 + loc] = acc11[row] + bv1;
  }
}

extern "C" void kernel_launch(void* const* d_in, const int* in_sizes, int n_in,
                              void* d_out, int out_size, void* d_ws, size_t ws_size,
                              hipStream_t stream) {
  (void)in_sizes; (void)n_in; (void)out_size; (void)d_ws; (void)ws_size;
  const float* x    = (const float*)d_in[0];
  const float* wgt  = (const float*)d_in[1];
  const float* bias = (const float*)d_in[2];
  float* out = (float*)d_out;
  // 4096 spatial locations, one wave32 per location.
  lc2d_wmma_f32_kernel<<<dim3(64 * 64), dim3(32), 0, stream>>>(x, wgt, bias, out);
}